// ObjectSegmentation_81338090651858
// MI455X (gfx1250) — compile-verified
//
#include <hip/hip_runtime.h>
#include <hip/hip_bf16.h>

#define NPROP 2000
#define NCLS  81
#define MROW  63504   // 28*28*81 floats per mask row
#define MROW4 15876   // MROW / 4 (float4 units)
#define PADN  300
#define NT16  125     // 2000 / 16 exact
#define SORTN 2048

typedef float v8f __attribute__((ext_vector_type(8)));
typedef float v2f __attribute__((ext_vector_type(2)));

// ---------------------------------------------------------------- kernel 1
// Per-proposal: decode top-class box, clip, max score over classes 1..80.
__global__ void k_boxes(const float* __restrict__ meta,
                        const float* __restrict__ deltas,
                        const float* __restrict__ props,
                        const float* __restrict__ scores,
                        float* __restrict__ oboxes,
                        float* __restrict__ omax) {
    int i = blockIdx.x * blockDim.x + threadIdx.x;
    if (i >= NPROP) return;
    float h = meta[0], w = meta[1];
    float inv = 1.0f / meta[2];
    float x1 = props[i * 4 + 0] * inv, y1 = props[i * 4 + 1] * inv;
    float x2 = props[i * 4 + 2] * inv, y2 = props[i * 4 + 3] * inv;
    float a = x2 - x1, b = y2 - y1;
    float cx = x1 + 0.5f * a, cy = y1 + 0.5f * b;

    const float* sc = scores + (size_t)i * NCLS;
    int   top  = 0;
    float best = sc[0];
    float msc  = -__builtin_inff();
    for (int c = 1; c < NCLS; ++c) {
        float v = sc[c];
        if (v > best) { best = v; top = c; }   // first-occurrence argmax
        msc = fmaxf(msc, v);                    // max over classes 1..80
    }

    const float* d = deltas + (size_t)i * 4 * NCLS + 4 * top;
    float dx = d[0], dy = d[1], dw = d[2], dh = d[3];
    float pcx = dx * a + cx, pcy = dy * b + cy;
    float pw  = expf(dw) * a, ph = expf(dh) * b;
    float bx1 = pcx - 0.5f * pw, by1 = pcy - 0.5f * ph;
    float bx2 = pcx + 0.5f * pw, by2 = pcy + 0.5f * ph;
    bx1 = fminf(fmaxf(bx1, 0.f), w - 1.f);
    by1 = fminf(fmaxf(by1, 0.f), h - 1.f);
    bx2 = fminf(fmaxf(bx2, 0.f), w - 1.f);
    by2 = fminf(fmaxf(by2, 0.f), h - 1.f);
    oboxes[i * 4 + 0] = bx1; oboxes[i * 4 + 1] = by1;
    oboxes[i * 4 + 2] = bx2; oboxes[i * 4 + 3] = by2;
    omax[i] = msc;
}

// ---------------------------------------------------------------- kernel 2
// Single-block bitonic sort: ascending by -score == descending by score.
__global__ __launch_bounds__(1024) void k_sort(const float* __restrict__ maxsc,
                                               const float* __restrict__ boxes,
                                               int* __restrict__ order,
                                               float* __restrict__ sboxes,
                                               float* __restrict__ sarea) {
    __shared__ float key[SORTN];
    __shared__ int   val[SORTN];
    int t = threadIdx.x;
    for (int idx = t; idx < SORTN; idx += 1024) {
        key[idx] = (idx < NPROP) ? -maxsc[idx] : __builtin_inff();
        val[idx] = idx;
    }
    __syncthreads();
    for (int k = 2; k <= SORTN; k <<= 1) {
        for (int j = k >> 1; j > 0; j >>= 1) {
            for (int idx = t; idx < SORTN; idx += 1024) {
                int ixj = idx ^ j;
                if (ixj > idx) {
                    bool up = ((idx & k) == 0);
                    float ka = key[idx], kb = key[ixj];
                    if ((ka > kb) == up) {
                        key[idx] = kb; key[ixj] = ka;
                        int va = val[idx]; val[idx] = val[ixj]; val[ixj] = va;
                    }
                }
            }
            __syncthreads();
        }
    }
    for (int idx = t; idx < NPROP; idx += 1024) {
        int o = val[idx];
        order[idx] = o;
        float bx1 = boxes[o * 4 + 0], by1 = boxes[o * 4 + 1];
        float bx2 = boxes[o * 4 + 2], by2 = boxes[o * 4 + 3];
        sboxes[idx * 4 + 0] = bx1; sboxes[idx * 4 + 1] = by1;
        sboxes[idx * 4 + 2] = bx2; sboxes[idx * 4 + 3] = by2;
        sarea[idx] = fmaxf(bx2 - bx1, 0.f) * fmaxf(by2 - by1, 0.f);
    }
}

// ---------------------------------------------------------------- kernel 3
// 16x16-tile IoU with WMMA: V_WMMA_F32_16X16X4_F32 computes the pairwise
// denominator term D[m,n] = area_row[m] + area_col[n] for the whole tile
// (A rows = [area_m, 1, 0, 0]; B cols = [1, area_n, 0, 0]^T).  Per-pair
// intersection stays in VALU; suppression bits packed via wave32 ballot.
__global__ __launch_bounds__(32) void k_iou_wmma(const float* __restrict__ sboxes,
                                                 const float* __restrict__ sarea,
                                                 unsigned short* __restrict__ mask16) {
    __shared__ float4 rb[16];          // row boxes of this tile
    int t  = threadIdx.x;
    int ti = blockIdx.x, tn = blockIdx.y;     // row tile, col tile
    int lo = t & 15;
    bool hi = (t >= 16);
    int r0 = ti * 16, n0 = tn * 16;

    const float4* sb4 = (const float4*)sboxes;
    if (!hi) rb[lo] = sb4[r0 + lo];           // lanes 0..15 stage row boxes
    float ra = sarea[r0 + lo];                // row areas (A, lanes 0..15)
    float ca = sarea[n0 + lo];                // col areas (B, lanes 0..15)
    float4 cbox = sb4[n0 + lo];               // this lane's column box
    __syncthreads();

    // f32 16x4 A layout: lanes 0-15 hold K=0 (v0) / K=1 (v1); lanes 16-31 K=2,3.
    // f32 4x16 B layout mirrors it.  K=2,3 contributions are zeroed.
    v2f A, B;
    A.x = hi ? 0.f : ra;   A.y = hi ? 0.f : 1.f;
    B.x = hi ? 0.f : 1.f;  B.y = hi ? 0.f : ca;
    v8f c = {};
    c = __builtin_amdgcn_wmma_f32_16x16x4_f32(false, A, false, B,
                                              (short)0, c, false, false);
    // c[v] = area_row[v + 8*hi] + area_col[lo]   (C/D layout)

    int jj = n0 + lo;                         // this lane's sorted column index
#pragma unroll
    for (int v = 0; v < 8; ++v) {
        int m  = v + (hi ? 8 : 0);
        int rr = r0 + m;
        float4 rbox = rb[m];
        float ix1 = fmaxf(rbox.x, cbox.x);
        float iy1 = fmaxf(rbox.y, cbox.y);
        float ix2 = fminf(rbox.z, cbox.z);
        float iy2 = fminf(rbox.w, cbox.w);
        float inter = fmaxf(ix2 - ix1, 0.f) * fmaxf(iy2 - iy1, 0.f);
        float iou   = inter / fmaxf(c[v] - inter, 1e-8f);
        bool pred = (jj > rr) && (iou > 0.5f);
        unsigned b = __builtin_amdgcn_ballot_w32(pred);
        if (t == 0) {
            // low 16 ballot bits: row v, cols n0..n0+15; high 16: row v+8
            mask16[(size_t)(r0 + v)     * NT16 + tn] = (unsigned short)(b & 0xffffu);
            mask16[(size_t)(r0 + v + 8) * NT16 + tn] = (unsigned short)(b >> 16);
        }
    }
}

// ---------------------------------------------------------------- kernel 4
// Sequential greedy reduce over the 16-bit suppression words. One wave32 —
// LDS ops within a single wave are ordered, so no barriers in the loop.
__global__ __launch_bounds__(32) void k_reduce(const unsigned short* __restrict__ mask16,
                                               const int* __restrict__ order,
                                               int* __restrict__ sel) {
    __shared__ unsigned short remv[128];
    int t = threadIdx.x;
    for (int w = t; w < 128; w += 32) remv[w] = 0;
    __syncthreads();
    int kcount = 0;
    for (int i = 0; i < NPROP; ++i) {
        unsigned rw = remv[i >> 4];
        if (!((rw >> (i & 15)) & 1u)) {
            const unsigned short* row = mask16 + (size_t)i * NT16;
            for (int w = t; w < NT16; w += 32) remv[w] |= row[w];
            if (kcount < PADN && t == 0) sel[kcount] = order[i];
            ++kcount;   // uniform across lanes
        }
    }
    for (int r = kcount + t; r < PADN; r += 32) sel[r] = -1;
}

// ---------------------------------------------------------------- kernel 5
// boxes_out (300x4) and scores_out (300x81); zeros for empty slots.
__global__ void k_small(const int* __restrict__ sel,
                        const float* __restrict__ boxes,
                        const float* __restrict__ scores,
                        float* __restrict__ out) {
    int r = blockIdx.x, t = threadIdx.x;
    int s = sel[r];
    if (t < 4) {
        out[r * 4 + t] = (s >= 0) ? boxes[s * 4 + t] : 0.f;
    } else if (t < 4 + NCLS) {
        int c = t - 4;
        out[(size_t)PADN * 4 + (size_t)r * NCLS + c] =
            (s >= 0) ? scores[(size_t)s * NCLS + c] : 0.f;
    }
}

// ---------------------------------------------------------------- kernel 6
// Bulk mask gather: one block per output row, streamed through a 64KB LDS
// staging buffer with CDNA5 async global<->LDS B128 transfers (ASYNCcnt).
__global__ __launch_bounds__(256) void k_masks(const int* __restrict__ sel,
                                               const float* __restrict__ masks,
                                               float* __restrict__ outm) {
    __shared__ float4 buf[4096];   // 64 KB staging
    int r = blockIdx.x, t = threadIdx.x;
    int s = sel[r];
    float4* dst = (float4*)outm + (size_t)r * MROW4;
    if (s < 0) {
        float4 z; z.x = z.y = z.z = z.w = 0.f;
        for (int idx = t; idx < MROW4; idx += 256) dst[idx] = z;
        return;
    }
    const float4* src = (const float4*)masks + (size_t)s * MROW4;
    // Low 32 bits of the generic shared-aperture address == wave-relative
    // LDS byte offset (aperture puts the LDS offset in addr[31:0]).
    unsigned ldsbase = (unsigned)(size_t)(&buf[0]);

    for (int base = 0; base < MROW4; base += 4096) {
        int count = MROW4 - base;
        if (count > 4096) count = 4096;
#pragma unroll
        for (int u = 0; u < 16; ++u) {
            int slot = u * 256 + t;
            if (slot < count) {
                unsigned loff = ldsbase + (unsigned)slot * 16u;
                unsigned long long ga = (unsigned long long)(src + base + slot);
                asm volatile("global_load_async_to_lds_b128 %0, %1, off"
                             :: "v"(loff), "v"(ga) : "memory");
            }
        }
        asm volatile("s_wait_asynccnt 0x0" ::: "memory");
#pragma unroll
        for (int u = 0; u < 16; ++u) {
            int slot = u * 256 + t;
            if (slot < count) {
                unsigned loff = ldsbase + (unsigned)slot * 16u;
                unsigned long long ga = (unsigned long long)(dst + base + slot);
                asm volatile("global_store_async_from_lds_b128 %0, %1, off"
                             :: "v"(ga), "v"(loff) : "memory");
            }
        }
        asm volatile("s_wait_asynccnt 0x0" ::: "memory");
    }
}

// ---------------------------------------------------------------- launch
extern "C" void kernel_launch(void* const* d_in, const int* in_sizes, int n_in,
                              void* d_out, int out_size, void* d_ws, size_t ws_size,
                              hipStream_t stream) {
    const float* meta   = (const float*)d_in[0];   // (1,3)
    const float* deltas = (const float*)d_in[1];   // (1,2000,324)
    const float* props  = (const float*)d_in[2];   // (1,2000,4)
    const float* scores = (const float*)d_in[3];   // (1,2000,81)
    const float* masks  = (const float*)d_in[4];   // (1,2000,28,28,81)
    float* out = (float*)d_out;                    // boxes|scores|masks concat

    // workspace carve-up (~589 KB)
    float*          w_boxes  = (float*)d_ws;                       // 2000*4 f
    float*          w_maxsc  = w_boxes + NPROP * 4;                // 2000 f
    int*            w_order  = (int*)(w_maxsc + NPROP);            // 2000 i
    float*          w_sboxes = (float*)(w_order + NPROP);          // 2000*4 f (16B aligned)
    float*          w_sarea  = w_sboxes + NPROP * 4;               // 2000 f
    unsigned short* w_mask16 = (unsigned short*)(w_sarea + NPROP); // 2000*125 u16
    int*            w_sel    = (int*)(w_mask16 + (size_t)NPROP * NT16); // 300 i

    k_boxes<<<(NPROP + 255) / 256, 256, 0, stream>>>(meta, deltas, props, scores,
                                                     w_boxes, w_maxsc);
    k_sort<<<1, 1024, 0, stream>>>(w_maxsc, w_boxes, w_order, w_sboxes, w_sarea);
    k_iou_wmma<<<dim3(NT16, NT16), 32, 0, stream>>>(w_sboxes, w_sarea, w_mask16);
    k_reduce<<<1, 32, 0, stream>>>(w_mask16, w_order, w_sel);
    k_small<<<PADN, 96, 0, stream>>>(w_sel, w_boxes, scores, out);
    k_masks<<<PADN, 256, 0, stream>>>(w_sel, masks,
                                      out + (size_t)PADN * 4 + (size_t)PADN * NCLS);
}